// algo_cmdnet_36936718745669
// MI455X (gfx1250) — compile-verified
//
#include <hip/hip_runtime.h>
#include <math.h>

// CMDNet forward for MI455X (gfx1250, wave32).
// One workgroup (256 threads = 8 waves) per batch element.
//   B=4096, Nr=128, Nt=64, M=16, NUM_ITER=64.
//
// Phase 1: stage Ht[b] (128x64 f32, 32KB) into LDS.
// Phase 2: HH = Ht^T Ht via V_WMMA_F32_16X16X4_F32 (each wave: 2 output
//          tiles x 32 chained WMMAs over K=128). yH = yt.Ht via shfl-reduce.
// Phase 3: 64 gradient layers fully on-chip. G[64x16] lives in registers
//          (4 floats per thread: thread t owns stream n=t>>2, M-slice q=t&3).
//          Softmax reductions over M=16 use __shfl_xor across the 4-lane
//          group (wave32 native); the 64x64 matvec reads HH from LDS.
// Phase 4: readout ft (B*64*16) then xt (B*64), concatenated in d_out.

#define NRR 128
#define NTT 64
#define MMM 16
#define ITERS 64

typedef __attribute__((ext_vector_type(2))) float v2f;
typedef __attribute__((ext_vector_type(8))) float v8f;

__global__ __launch_bounds__(256) void cmdnet_fused_kernel(
    const float* __restrict__ yt,     // [B, 128]
    const float* __restrict__ Ht,     // [B, 128, 64]
    const float* __restrict__ sig0,   // [B]
    const float* __restrict__ mconst, // [16]
    const float* __restrict__ alpha,  // [16]
    const float* __restrict__ taui,   // [65]
    const float* __restrict__ delta,  // [64]
    float* __restrict__ out_ft,       // [B, 64, 16]
    float* __restrict__ out_xt)       // [B, 64]
{
    __shared__ float HtS[NRR][NTT];        // 32 KB
    __shared__ float HHs[NTT][NTT + 1];    // 16.6 KB (padded)
    __shared__ float ytS[NRR];
    __shared__ float yHs[NTT];
    __shared__ float xts[NTT];
    __shared__ float xHHs[NTT];
    __shared__ float tauiS[ITERS + 1];
    __shared__ float deltaS[ITERS];

    const int b = blockIdx.x;
    const int t = threadIdx.x;

    // ---------------- Phase 1: stage inputs ----------------
    {
        const float4* src = (const float4*)(Ht + (size_t)b * NRR * NTT);
        float4* dst = (float4*)&HtS[0][0];
        #pragma unroll
        for (int i = 0; i < (NRR * NTT / 4) / 256; ++i)
            dst[t + 256 * i] = src[t + 256 * i];
    }
    if (t < NRR) ytS[t] = yt[(size_t)b * NRR + t];
    if (t < ITERS + 1) tauiS[t] = taui[t];
    if (t < ITERS) deltaS[t] = delta[t];
    __syncthreads();

    // ---------------- Phase 2a: yH = yt . Ht (matched filter) ----------------
    {
        const int k = t >> 2, q = t & 3;
        float acc = 0.f;
        #pragma unroll 8
        for (int r = q * 32; r < q * 32 + 32; ++r)
            acc += ytS[r] * HtS[r][k];
        acc += __shfl_xor(acc, 1, 32);
        acc += __shfl_xor(acc, 2, 32);
        if (q == 0) yHs[k] = acc;
    }

    // ---------------- Phase 2b: HH = Ht^T Ht via WMMA f32 16x16x4 ----------------
    {
        const int wave = t >> 5;
        const int lane = t & 31;
        const int half = lane >> 4;    // K split: lanes 0-15 -> K={0,1}, 16-31 -> K={2,3}
        const int ml   = lane & 15;    // M / N index within tile
        #pragma unroll
        for (int rep = 0; rep < 2; ++rep) {
            const int tile = wave * 2 + rep;          // 16 tiles total (4x4)
            const int I = (tile >> 2) * 16;           // output rows
            const int J = (tile & 3) * 16;            // output cols
            v8f c = {};
            #pragma unroll 4
            for (int k = 0; k < NRR; k += 4) {
                v2f a, bm;
                // A = Ht[k..k+3, I..I+15]^T  (16x4, M=ml)
                a.x  = HtS[k + half * 2 + 0][I + ml];
                a.y  = HtS[k + half * 2 + 1][I + ml];
                // B = Ht[k..k+3, J..J+15]    (4x16, N=ml)
                bm.x = HtS[k + half * 2 + 0][J + ml];
                bm.y = HtS[k + half * 2 + 1][J + ml];
                c = __builtin_amdgcn_wmma_f32_16x16x4_f32(
                        false, a, false, bm, (short)0, c, false, false);
            }
            // C/D layout: VGPR g, lanes 0-15 -> M=g, lanes 16-31 -> M=8+g, N=lane&15
            #pragma unroll
            for (int g = 0; g < 8; ++g)
                HHs[I + g + half * 8][J + ml] = c[g];
        }
    }
    __syncthreads();

    // ---------------- Phase 3: 64 gradient-descent layers ----------------
    const int n = t >> 2;   // stream index 0..63
    const int q = t & 3;    // owns M entries 4q..4q+3

    float G0 = 0.f, G1 = 0.f, G2 = 0.f, G3 = 0.f;
    float m0 = mconst[4 * q + 0], m1 = mconst[4 * q + 1];
    float m2 = mconst[4 * q + 2], m3 = mconst[4 * q + 3];
    float la0 = __logf(alpha[4 * q + 0]), la1 = __logf(alpha[4 * q + 1]);
    float la2 = __logf(alpha[4 * q + 2]), la3 = __logf(alpha[4 * q + 3]);
    const float s0 = sig0[b];
    const float sig2 = s0 * s0;

    float f0, f1, f2, f3;  // softmax probabilities (this thread's 4 of 16)

    for (int it = 0; it <= ITERS; ++it) {
        const float ta = fabsf(tauiS[it]);
        const float scale = (it == 0) ? 1.0f : ta;

        // softmax over M=16 (4 lanes x 4 regs), max-subtracted like jax
        float z0 = (la0 + G0) * scale, z1 = (la1 + G1) * scale;
        float z2 = (la2 + G2) * scale, z3 = (la3 + G3) * scale;
        float mx = fmaxf(fmaxf(z0, z1), fmaxf(z2, z3));
        mx = fmaxf(mx, __shfl_xor(mx, 1, 32));
        mx = fmaxf(mx, __shfl_xor(mx, 2, 32));
        z0 = __expf(z0 - mx); z1 = __expf(z1 - mx);
        z2 = __expf(z2 - mx); z3 = __expf(z3 - mx);
        float s = z0 + z1 + z2 + z3;
        s += __shfl_xor(s, 1, 32);
        s += __shfl_xor(s, 2, 32);
        const float inv = 1.0f / s;
        f0 = z0 * inv; f1 = z1 * inv; f2 = z2 * inv; f3 = z3 * inv;

        // xt[n] = sum_m ft*m
        float xp = f0 * m0 + f1 * m1 + f2 * m2 + f3 * m3;
        xp += __shfl_xor(xp, 1, 32);
        xp += __shfl_xor(xp, 2, 32);
        if (q == 0) xts[n] = xp;

        if (it == ITERS) break;  // final layer: readout only
        __syncthreads();

        // xHH[k] = sum_n xt[n] * HH[n][k]   (k = n here; 16 MACs/thread)
        float acc = 0.f;
        #pragma unroll 4
        for (int nn = q * 16; nn < q * 16 + 16; ++nn)
            acc += xts[nn] * HHs[nn][n];
        acc += __shfl_xor(acc, 1, 32);
        acc += __shfl_xor(acc, 2, 32);
        if (q == 0) xHHs[n] = acc;
        __syncthreads();

        // G -= delta * ( sig2*(1-exp(-G)) + ta*ft*(m - xt)*(xHH - yH) )
        const float rr = (xHHs[n] - yHs[n]) * ta;
        const float xtn = xts[n];
        const float d = deltaS[it];
        G0 -= d * (sig2 * (1.f - __expf(-G0)) + f0 * (m0 - xtn) * rr);
        G1 -= d * (sig2 * (1.f - __expf(-G1)) + f1 * (m1 - xtn) * rr);
        G2 -= d * (sig2 * (1.f - __expf(-G2)) + f2 * (m2 - xtn) * rr);
        G3 -= d * (sig2 * (1.f - __expf(-G3)) + f3 * (m3 - xtn) * rr);
        __syncthreads();  // protect xts/xHHs before next layer rewrites them
    }

    // ---------------- Phase 4: readout ----------------
    {
        float4 fv = make_float4(f0, f1, f2, f3);
        float4* dst = (float4*)(out_ft + (size_t)b * NTT * MMM + n * MMM + 4 * q);
        *dst = fv;
        if (q == 0) out_xt[(size_t)b * NTT + n] = xts[n];
    }
}

extern "C" void kernel_launch(void* const* d_in, const int* in_sizes, int n_in,
                              void* d_out, int out_size, void* d_ws, size_t ws_size,
                              hipStream_t stream) {
    const float* yt    = (const float*)d_in[0];
    const float* Ht    = (const float*)d_in[1];
    const float* sig0  = (const float*)d_in[2];
    const float* m     = (const float*)d_in[3];
    const float* alpha = (const float*)d_in[4];
    const float* taui  = (const float*)d_in[5];
    const float* delta = (const float*)d_in[6];

    const int B = in_sizes[2];  // sigmat0 has one entry per batch element

    float* out_ft = (float*)d_out;                        // [B,64,16] first
    float* out_xt = out_ft + (size_t)B * NTT * MMM;       // then [B,64]

    cmdnet_fused_kernel<<<B, 256, 0, stream>>>(
        yt, Ht, sig0, m, alpha, taui, delta, out_ft, out_xt);
}